// ChannelMerger_68126771249539
// MI455X (gfx1250) — compile-verified
//
#include <hip/hip_runtime.h>
#include <math.h>

typedef __attribute__((ext_vector_type(2))) float v2f;
typedef __attribute__((ext_vector_type(8))) float v8f;

#define B_BATCH 32
#define T_LEN   8192
#define C_CH    128
#define O_OUT   64
#define PDIM    288
#define NFREQ   12
#define NLOC    144

// ---------------------------------------------------------------------------
// Kernel 1: scores[b,o,c] = dot(heads[o,:], fourier_emb(positions[b,c,:]))
// One 64-thread block per (b,c). cos/sin table built cooperatively in LDS.
// ---------------------------------------------------------------------------
__global__ __launch_bounds__(64) void cm_scores(
    const float* __restrict__ positions,  // [B,C,2]
    const float* __restrict__ heads,      // [O,288]
    float* __restrict__ scores)           // [B,O,C]
{
    __shared__ float cs[PDIM];
    const int c   = blockIdx.x;
    const int b   = blockIdx.y;
    const int tid = threadIdx.x;

    const float margin = 0.2f;
    const float scale  = 2.0f * 3.14159265358979323846f / (1.0f + 2.0f * margin);
    const float px = (positions[((size_t)b * C_CH + c) * 2 + 0] + margin) * scale;
    const float py = (positions[((size_t)b * C_CH + c) * 2 + 1] + margin) * scale;

    for (int idx = tid; idx < NLOC; idx += 64) {
        const int i = idx / NFREQ;
        const int j = idx - i * NFREQ;
        const float loc = px * (float)i + py * (float)j;
        float s, co;
        __sincosf(loc, &s, &co);
        cs[idx]        = co;   // cos half
        cs[NLOC + idx] = s;    // sin half
    }
    __syncthreads();

    // one head per thread (O == 64 == blockDim.x)
    const int o = tid;
    const float4* __restrict__ hrow = (const float4*)(heads + (size_t)o * PDIM);
    const float4* __restrict__ csv  = (const float4*)cs;
    float acc = 0.0f;
#pragma unroll 4
    for (int d = 0; d < PDIM / 4; ++d) {
        const float4 h = hrow[d];
        const float4 v = csv[d];
        acc += h.x * v.x + h.y * v.y + h.z * v.z + h.w * v.w;
    }
    scores[((size_t)b * O_OUT + o) * C_CH + c] = acc;
}

// ---------------------------------------------------------------------------
// Kernel 2: in-place softmax over C (=128) for each (b,o) row.
// ---------------------------------------------------------------------------
__global__ __launch_bounds__(128) void cm_softmax(float* __restrict__ sw)  // [B,O,C]
{
    __shared__ float red[128];
    const int o   = blockIdx.x;
    const int b   = blockIdx.y;
    const int tid = threadIdx.x;

    float* __restrict__ row = sw + ((size_t)b * O_OUT + o) * C_CH;
    const float v = row[tid];

    red[tid] = v;
    __syncthreads();
#pragma unroll
    for (int s = 64; s > 0; s >>= 1) {
        if (tid < s) red[tid] = fmaxf(red[tid], red[tid + s]);
        __syncthreads();
    }
    const float m = red[0];
    __syncthreads();

    const float e = __expf(v - m);
    red[tid] = e;
    __syncthreads();
#pragma unroll
    for (int s = 64; s > 0; s >>= 1) {
        if (tid < s) red[tid] += red[tid + s];
        __syncthreads();
    }
    row[tid] = e * (1.0f / red[0]);
}

// ---------------------------------------------------------------------------
// Kernel 3: out[b,t,o] = sum_c eeg[b,t,c] * w[b,o,c]
// GEMM per batch: M=T=8192, N=O=64, K=C=128 using V_WMMA_F32_16X16X4_F32.
// Grid (T/128, B); 256 threads = 8 waves; each wave owns a 16x64 output strip.
// ---------------------------------------------------------------------------
__device__ __forceinline__ v8f wmma4(v2f a, v2f b, v8f c) {
    // (neg_a, A, neg_b, B, c_mod, C, reuse_a, reuse_b)
    return __builtin_amdgcn_wmma_f32_16x16x4_f32(false, a, false, b, (short)0, c,
                                                 false, false);
}

__global__ __launch_bounds__(256) void cm_gemm(
    const float* __restrict__ eeg,  // [B,T,C]
    const float* __restrict__ w,    // [B,O,C] (softmaxed weights)
    float* __restrict__ out)        // [B,T,O]
{
    __shared__ float wl[O_OUT * C_CH];  // 32 KB: full weight matrix for batch b
    const int b   = blockIdx.y;
    const int tid = threadIdx.x;

    // Stage weights into LDS (float4 / b128 coalesced)
    {
        const float4* __restrict__ src = (const float4*)(w + (size_t)b * O_OUT * C_CH);
        float4* dst = (float4*)wl;
#pragma unroll
        for (int i = 0; i < (O_OUT * C_CH / 4) / 256; ++i)
            dst[tid + i * 256] = src[tid + i * 256];
    }
    __syncthreads();

    const int wave  = tid >> 5;
    const int lane  = tid & 31;
    const int lhalf = lane >> 4;   // 0: K pair {k,k+1}; 1: {k+2,k+3}
    const int lmod  = lane & 15;   // A: row within tile / B: output column
    const int t0    = blockIdx.x * 128 + wave * 16;

    // A fragment (16x4 f32): lane<16 -> M=lane holds K=k,k+1; lane>=16 -> K=k+2,k+3
    const float* __restrict__ aptr =
        eeg + ((size_t)b * T_LEN + t0 + lmod) * C_CH + lhalf * 2;
    // B fragment (4x16 f32), mirrored layout over N: lane<16 -> N=lmod, K=k,k+1
    const float* __restrict__ bbase = wl + (size_t)lmod * C_CH + lhalf * 2;

    v8f acc0 = {}, acc1 = {}, acc2 = {}, acc3 = {};

#pragma unroll 4
    for (int k = 0; k < C_CH; k += 4) {
        const v2f a  = *(const v2f*)(aptr + k);
        const v2f b0 = *(const v2f*)(bbase + 0 * 16 * C_CH + k);
        const v2f b1 = *(const v2f*)(bbase + 1 * 16 * C_CH + k);
        const v2f b2 = *(const v2f*)(bbase + 2 * 16 * C_CH + k);
        const v2f b3 = *(const v2f*)(bbase + 3 * 16 * C_CH + k);
        acc0 = wmma4(a, b0, acc0);
        acc1 = wmma4(a, b1, acc1);
        acc2 = wmma4(a, b2, acc2);
        acc3 = wmma4(a, b3, acc3);
    }

    // D layout (16x16 f32): lanes 0-15 -> N=lane, M=v (rows 0-7);
    //                       lanes 16-31 -> N=lane-16, M=8+v.
    float* __restrict__ obase =
        out + ((size_t)b * T_LEN + t0 + lhalf * 8) * O_OUT + lmod;
#pragma unroll
    for (int v = 0; v < 8; ++v) {
        obase[(size_t)v * O_OUT + 0]  = acc0[v];
        obase[(size_t)v * O_OUT + 16] = acc1[v];
        obase[(size_t)v * O_OUT + 32] = acc2[v];
        obase[(size_t)v * O_OUT + 48] = acc3[v];
    }
}

// ---------------------------------------------------------------------------
extern "C" void kernel_launch(void* const* d_in, const int* in_sizes, int n_in,
                              void* d_out, int out_size, void* d_ws, size_t ws_size,
                              hipStream_t stream) {
    const float* eeg       = (const float*)d_in[0];  // [B,T,C]
    const float* positions = (const float*)d_in[1];  // [B,C,2]
    const float* heads     = (const float*)d_in[2];  // [O,288]
    // d_in[3] (sub) unused by the reference
    float* sw  = (float*)d_ws;   // [B,O,C] scores -> weights (1 MB)
    float* o   = (float*)d_out;  // [B,T,O]

    cm_scores <<<dim3(C_CH, B_BATCH),      64, 0, stream>>>(positions, heads, sw);
    cm_softmax<<<dim3(O_OUT, B_BATCH),    128, 0, stream>>>(sw);
    cm_gemm   <<<dim3(T_LEN / 128, B_BATCH), 256, 0, stream>>>(eeg, sw, o);
}